// SinenetLayerIndiv_76433238000018
// MI455X (gfx1250) — compile-verified
//
#include <hip/hip_runtime.h>
#include <hip/hip_bf16.h>
#include <math.h>

// SinenetLayerIndiv on gfx1250 (MI455X).
// Phase 1: harmonic-rotation recurrence -> P/Q partial sums  (VALU, lane-per-harmonic)
// Phase 2: sparse projection as f32 GEMM via V_WMMA_F32_16X16X4_F32
//
// Shapes: S=8, B=64 (SB=512), T=640, D=128, harmonics I = D/4 = 32.

#define TIME_LEN   640
#define OUTPUT_DIM 128
#define SB_TOTAL   512
#define T_WAV      (1.0f / 16000.0f)
#define LOG_F_MEAN 5.02654f
#define LOG_F_STD  0.373288f
#define RESEED     160   // exact sincos re-seed interval (bounds rotation drift)

typedef __attribute__((ext_vector_type(2))) float v2f;
typedef __attribute__((ext_vector_type(8))) float v8f;

// ---------------------------------------------------------------------------
// Phase 1: one wave per (s,b); lane i (0..31) owns harmonic i.
//   theta_t = f * (t*T_WAV - tau);   P_i = sum_t sin(i*theta_t)*x_t, Q_i = cos.
// sin/cos of i*theta advanced by a fixed rotation delta_i = i*f*T_WAV.
// Output PQ[sb][2i]=P_i, PQ[sb][2i+1]=Q_i  -> coalesced b64 store per lane.
// ---------------------------------------------------------------------------
__global__ __launch_bounds__(256) void sinenet_pq_kernel(
    const float* __restrict__ x,    // [SB, T]
    const float* __restrict__ nlf,  // [SB]
    const float* __restrict__ tau,  // [SB]
    float* __restrict__ pq)         // [SB, 64]
{
    const int wave = threadIdx.x >> 5;       // 8 waves / block
    const int lane = threadIdx.x & 31;       // harmonic index i
    const int sb   = blockIdx.x * 8 + wave;  // 64 blocks -> 512 (s,b)

    const float f  = expf(nlf[sb] * LOG_F_STD + LOG_F_MEAN);
    const float tv = tau[sb];
    const float fi = f * (float)lane;        // i * f

    float sd, cd;                            // rotation by delta_i per sample
    sincosf(fi * T_WAV, &sd, &cd);

    float P = 0.0f, Q = 0.0f;
    const float* xp = x + sb * TIME_LEN;

    for (int t0 = 0; t0 < TIME_LEN; t0 += RESEED) {
        float s, c;                          // exact re-seed at t0
        sincosf(fi * ((float)t0 * T_WAV - tv), &s, &c);
#pragma unroll 4
        for (int t = 0; t < RESEED; ++t) {
            const float xt = xp[t0 + t];     // wave-uniform load (L0/L2 hit)
            P = fmaf(s, xt, P);
            Q = fmaf(c, xt, Q);
            const float sn = fmaf(s, cd,  c * sd);
            const float cn = fmaf(c, cd, -(s * sd));
            s = sn; c = cn;
        }
    }

    pq[sb * 64 + 2 * lane]     = P;
    pq[sb * 64 + 2 * lane + 1] = Q;
}

// ---------------------------------------------------------------------------
// Phase 2: h[512,128] = PQ[512,64] x W[64,128] with
//   W[2i, d]   = a_d * cos(phi_d) * (i == d/4)
//   W[2i+1, d] = a_d * sin(phi_d) * (i == d/4)
// Column tile tn covers d in [16tn,16tn+16) -> nonzero k's in [8tn, 8tn+8):
// exactly two K=4 WMMA steps per 16x16 output tile.
//
// Fragment layouts (ISA 7.12.2):
//   A 16x4 f32 : lane L holds M=L%16, K = 2*(L/16) + {0,1}     (2 VGPRs)
//   B  4x16 f32: lane L holds N=L%16, K = 2*(L/16) + {0,1}     (mirrors A)
//   C 16x16 f32: VGPR r -> M = r + 8*(L/16), N = L%16          (8 VGPRs)
// One tile per wave; 32 blocks x 8 waves = 256 tiles = 512 WMMAs.
// No divergent branches -> EXEC all ones (WMMA requirement).
// ---------------------------------------------------------------------------
__global__ __launch_bounds__(256) void sinenet_wmma_kernel(
    const float* __restrict__ pq,   // [SB, 64]
    const float* __restrict__ a,    // [D]
    const float* __restrict__ phi,  // [D]
    float* __restrict__ out)        // [SB, D]
{
    const int wave = threadIdx.x >> 5;
    const int lane = threadIdx.x & 31;
    const int tile = blockIdx.x * 8 + wave;  // 0..255
    const int tm   = tile >> 3;              // row tile 0..31  (16 sb rows)
    const int tn   = tile & 7;               // col tile 0..7   (16 d cols)
    const int half = lane >> 4;
    const int l16  = lane & 15;

    // --- A fragments: PQ rows ---
    const int row = tm * 16 + l16;
    const float* pqr = pq + row * 64;
    const int kA = tn * 8 + 2 * half;        // first WMMA k-pair base
    v2f A0 = { pqr[kA],     pqr[kA + 1] };
    v2f A1 = { pqr[kA + 4], pqr[kA + 5] };

    // --- B fragments: synthesize sparse W on the fly ---
    const int col = tn * 16 + l16;
    float sph, cph;
    sincosf(phi[col], &sph, &cph);
    const float av  = a[col];
    const float wc  = av * cph;              // W[2*ic,   col]
    const float wsn = av * sph;              // W[2*ic+1, col]
    const int ic2   = (col >> 2) << 1;       // 2*(col/4) in [8tn, 8tn+6]
    const bool h0 = (kA == ic2);             // pair {kA, kA+1} hits {2ic, 2ic+1}
    const bool h1 = (kA + 4 == ic2);
    v2f B0 = { h0 ? wc : 0.0f, h0 ? wsn : 0.0f };
    v2f B1 = { h1 ? wc : 0.0f, h1 ? wsn : 0.0f };

    v8f C = {};
    C = __builtin_amdgcn_wmma_f32_16x16x4_f32(false, A0, false, B0,
                                              (short)0, C, false, false);
    C = __builtin_amdgcn_wmma_f32_16x16x4_f32(false, A1, false, B1,
                                              (short)0, C, false, false);

    // --- store C: VGPR r -> row (tm*16 + r + 8*half), col ---
    float* orow = out + (tm * 16 + 8 * half) * OUTPUT_DIM + col;
#pragma unroll
    for (int r = 0; r < 8; ++r)
        orow[r * OUTPUT_DIM] = C[r];
}

// ---------------------------------------------------------------------------
extern "C" void kernel_launch(void* const* d_in, const int* in_sizes, int n_in,
                              void* d_out, int out_size, void* d_ws, size_t ws_size,
                              hipStream_t stream) {
    (void)in_sizes; (void)n_in; (void)out_size; (void)ws_size;
    const float* x   = (const float*)d_in[0];  // [8,64,640]
    const float* nlf = (const float*)d_in[1];  // [8,64,1]
    const float* tau = (const float*)d_in[2];  // [8,64,1]
    const float* a   = (const float*)d_in[3];  // [128]
    const float* phi = (const float*)d_in[4];  // [128]
    float* pq  = (float*)d_ws;                 // 512*64 f32 = 128 KB scratch
    float* out = (float*)d_out;                // 512*128 f32

    sinenet_pq_kernel<<<64, 256, 0, stream>>>(x, nlf, tau, pq);
    sinenet_wmma_kernel<<<32, 256, 0, stream>>>(pq, a, phi, out);
}